// SAGE_68582037782752
// MI455X (gfx1250) — compile-verified
//
#include <hip/hip_runtime.h>
#include <hip/hip_bf16.h>

// ---------------------------------------------------------------------------
// GraphSAGE (2x SAGEConv mean + link-pred MLP) for MI455X / gfx1250.
// Dense math runs on V_WMMA_F32_16X16X4_F32 (exact f32, matches reference).
// N=50000 (=16*3125), F=128, E=500000, P=200000 (=16*12500).
// ---------------------------------------------------------------------------

#define N_NODES 50000
#define FDIM    128
#define N_EDGES 500000
#define N_PAIRS 200000

typedef __attribute__((ext_vector_type(2))) float v2f;
typedef __attribute__((ext_vector_type(8))) float v8f;

// ---------------------------------------------------------------- utilities
__global__ void zero_f32(float* __restrict__ p, int n) {
    int i = blockIdx.x * blockDim.x + threadIdx.x;
    if (i < n) p[i] = 0.0f;
}

// cnt[dst] += 1 per edge
__global__ void count_edges(const int* __restrict__ dst, float* __restrict__ cnt, int e) {
    int i = blockIdx.x * blockDim.x + threadIdx.x;
    if (i < e) atomicAdd(&cnt[dst[i]], 1.0f);
}

// cnt -> 1/max(cnt,1)
__global__ void invert_cnt(float* __restrict__ cnt, int n) {
    int i = blockIdx.x * blockDim.x + threadIdx.x;
    if (i < n) {
        float c = cnt[i];
        cnt[i] = 1.0f / (c > 1.0f ? c : 1.0f);
    }
}

// agg[dst] += feat[src]; one lane handles 4 consecutive features (float4 load)
__global__ void scatter_add(const float* __restrict__ feat,
                            const int* __restrict__ src,
                            const int* __restrict__ dst,
                            float* __restrict__ agg, int e) {
    int t = blockIdx.x * blockDim.x + threadIdx.x;       // e * 32 work items
    int edge = t >> 5;
    if (edge >= e) return;
    int f0 = (t & 31) << 2;                              // 0,4,...,124
    const float4 v = *(const float4*)&feat[(size_t)src[edge] * FDIM + f0];
    float* a = &agg[(size_t)dst[edge] * FDIM + f0];
    atomicAdd(a + 0, v.x);
    atomicAdd(a + 1, v.y);
    atomicAdd(a + 2, v.z);
    atomicAdd(a + 3, v.w);
}

// ---------------------------------------------------------------- SAGE GEMM
// out[m,n] = relu( sum_k A1[m,k]*rs[m]*W1[n,k] + A2[m,k]*W2[n,k] + bias[n] )
// One block = 16-row strip; 8 waves cover the 8 N-tiles; A tiles staged in LDS.
__global__ __launch_bounds__(256)
void sage_gemm(const float* __restrict__ A1, const float* __restrict__ rowscale,
               const float* __restrict__ A2,
               const float* __restrict__ W1, const float* __restrict__ W2,
               const float* __restrict__ bias, float* __restrict__ out) {
    __shared__ float lA1[16 * FDIM];
    __shared__ float lA2[16 * FDIM];
    const int m0 = blockIdx.x * 16;

    // cooperative stage of both 16x128 A tiles (mean-divide fused via rowscale)
    for (int i = threadIdx.x; i < 16 * FDIM; i += 256) {
        int r = i >> 7;
        lA1[i] = A1[(size_t)(m0 + r) * FDIM + (i & 127)] * rowscale[m0 + r];
        lA2[i] = A2[(size_t)(m0 + r) * FDIM + (i & 127)];
    }
    __syncthreads();

    const int wave  = threadIdx.x >> 5;        // N-tile id (0..7)
    const int lane  = threadIdx.x & 31;
    const int n0    = wave * 16;
    const int row   = lane & 15;               // A: M index / B: N index
    const int khalf = (lane >> 4) * 2;         // K sub-offset (0 or 2)

    v8f acc = {};
    for (int k0 = 0; k0 < FDIM; k0 += 4) {
        int k = k0 + khalf;
        v2f a1 = *(const v2f*)&lA1[row * FDIM + k];
        v2f a2 = *(const v2f*)&lA2[row * FDIM + k];
        v2f b1 = *(const v2f*)&W1[(n0 + row) * FDIM + k];   // B[k][n] = W[n][k]
        v2f b2 = *(const v2f*)&W2[(n0 + row) * FDIM + k];
        acc = __builtin_amdgcn_wmma_f32_16x16x4_f32(false, a1, false, b1,
                                                    (short)0, acc, false, false);
        acc = __builtin_amdgcn_wmma_f32_16x16x4_f32(false, a2, false, b2,
                                                    (short)0, acc, false, false);
    }

    // D layout: vgpr i -> lanes 0-15: (M=i, N=lane); lanes 16-31: (M=8+i, N=lane-16)
    const int nn    = n0 + (lane & 15);
    const float bn  = bias[nn];
    const int mbase = m0 + ((lane >> 4) << 3);
#pragma unroll
    for (int i = 0; i < 8; ++i) {
        float v = acc[i] + bn;
        out[(size_t)(mbase + i) * FDIM + nn] = v > 0.0f ? v : 0.0f;
    }
}

// ------------------------------------------------------------- pair-MLP head
// out[p] = sigmoid( relu(concat(h2[a],h2[b]) @ W3^T + b3) @ W4^T + b4 )
// One block = 16 pairs; 8 waves cover the 128 hidden dims (K = 256 via WMMA).
__global__ __launch_bounds__(256)
void pair_mlp(const float* __restrict__ h2, const int* __restrict__ pairs,
              const float* __restrict__ W3, const float* __restrict__ b3,
              const float* __restrict__ W4, const float* __restrict__ b4,
              float* __restrict__ out) {
    __shared__ float z[16 * 256];        // 16 KB gathered concat features
    __shared__ float sred[16];
    const int p0 = blockIdx.x * 16;

    for (int i = threadIdx.x; i < 16 * 256; i += 256) {
        int r = i >> 8, k = i & 255;
        int node = pairs[(p0 + r) * 2 + (k >> 7)];
        z[i] = h2[(size_t)node * FDIM + (k & 127)];
    }
    if (threadIdx.x < 16) sred[threadIdx.x] = 0.0f;
    __syncthreads();

    const int wave  = threadIdx.x >> 5;
    const int lane  = threadIdx.x & 31;
    const int n0    = wave * 16;
    const int row   = lane & 15;
    const int khalf = (lane >> 4) * 2;

    v8f acc = {};
    for (int k0 = 0; k0 < 256; k0 += 4) {
        int k = k0 + khalf;
        v2f a = *(const v2f*)&z[row * 256 + k];
        v2f b = *(const v2f*)&W3[(n0 + row) * 256 + k];
        acc = __builtin_amdgcn_wmma_f32_16x16x4_f32(false, a, false, b,
                                                    (short)0, acc, false, false);
    }

    const int nn   = n0 + (lane & 15);
    const float bn = b3[nn];
    const float wn = W4[nn];
    float part[8];
#pragma unroll
    for (int i = 0; i < 8; ++i) {
        float v = acc[i] + bn;
        v = v > 0.0f ? v : 0.0f;
        part[i] = v * wn;                        // fuse the W4 dot product
    }
    // sum over the 16 lanes sharing each M row (lanes 0-15 / 16-31 separately)
#pragma unroll
    for (int off = 1; off < 16; off <<= 1)
#pragma unroll
        for (int i = 0; i < 8; ++i) part[i] += __shfl_xor(part[i], off, 32);

    if ((lane & 15) == 0) {                      // lanes 0 (M=0..7) and 16 (M=8..15)
        int mbase = (lane >> 4) << 3;
#pragma unroll
        for (int i = 0; i < 8; ++i) atomicAdd(&sred[mbase + i], part[i]);  // ds_add_f32
    }
    __syncthreads();
    if (threadIdx.x < 16) {
        float s = sred[threadIdx.x] + b4[0];
        out[p0 + threadIdx.x] = 1.0f / (1.0f + __expf(-s));
    }
}

// ------------------------------------------------------------------- driver
extern "C" void kernel_launch(void* const* d_in, const int* in_sizes, int n_in,
                              void* d_out, int out_size, void* d_ws, size_t ws_size,
                              hipStream_t stream) {
    const float* x    = (const float*)d_in[0];
    const int*   ei   = (const int*)d_in[1];
    const int*   prs  = (const int*)d_in[2];
    const float* Wl1  = (const float*)d_in[3];
    const float* Wr1  = (const float*)d_in[4];
    const float* b1   = (const float*)d_in[5];
    const float* Wl2  = (const float*)d_in[6];
    const float* Wr2  = (const float*)d_in[7];
    const float* b2   = (const float*)d_in[8];
    const float* W3   = (const float*)d_in[9];
    const float* b3   = (const float*)d_in[10];
    const float* W4   = (const float*)d_in[11];
    const float* b4   = (const float*)d_in[12];
    float* out = (float*)d_out;

    const int* src = ei;             // edge_index[0]
    const int* dst = ei + N_EDGES;   // edge_index[1]

    // workspace: agg | invcnt | h1 | h2   (~77 MB total)
    float* agg  = (float*)d_ws;
    float* invc = agg + (size_t)N_NODES * FDIM;
    float* h1   = invc + N_NODES;
    float* h2   = h1 + (size_t)N_NODES * FDIM;

    const int NF = N_NODES * FDIM;
    dim3 thr(256);

    // degree -> reciprocal (zero agg + cnt in one shot; they are contiguous)
    zero_f32<<<dim3((NF + N_NODES + 255) / 256), thr, 0, stream>>>(agg, NF + N_NODES);
    count_edges<<<dim3((N_EDGES + 255) / 256), thr, 0, stream>>>(dst, invc, N_EDGES);
    invert_cnt<<<dim3((N_NODES + 255) / 256), thr, 0, stream>>>(invc, N_NODES);

    // layer 1
    scatter_add<<<dim3((N_EDGES * 32) / 256), thr, 0, stream>>>(x, src, dst, agg, N_EDGES);
    sage_gemm<<<dim3(N_NODES / 16), thr, 0, stream>>>(agg, invc, x, Wl1, Wr1, b1, h1);

    // layer 2
    zero_f32<<<dim3((NF + 255) / 256), thr, 0, stream>>>(agg, NF);
    scatter_add<<<dim3((N_EDGES * 32) / 256), thr, 0, stream>>>(h1, src, dst, agg, N_EDGES);
    sage_gemm<<<dim3(N_NODES / 16), thr, 0, stream>>>(agg, invc, h1, Wl2, Wr2, b2, h2);

    // link-prediction head
    pair_mlp<<<dim3(N_PAIRS / 16), thr, 0, stream>>>(h2, prs, W3, b3, W4, b4, out);
}